// BioSimulatorHILO_32298154066728
// MI455X (gfx1250) — compile-verified
//
#include <hip/hip_runtime.h>
#include <cmath>

typedef __attribute__((ext_vector_type(2))) float v2f;
typedef __attribute__((ext_vector_type(8))) float v8f;

#define L2E 1.4426950408889634f

#if defined(__has_builtin) && __has_builtin(__builtin_amdgcn_exp2f)
#define FAST_EXP2(x) __builtin_amdgcn_exp2f(x)   // bare v_exp_f32
#else
#define FAST_EXP2(x) exp2f(x)
#endif

__device__ __forceinline__ float clampf(float v, float lo, float hi) {
    return fminf(fmaxf(v, lo), hi);
}

// ---------------------------------------------------------------------------
// Phase 1: per-electrode parameters.  2 batches x 1024 electrodes = 2048 thr.
// Writes to ws: [b][j][1024] with j = {0:k=1/(2*sigma_px^2), 1:Vx(px),
//                                      2:Vy(px), 3:log2(Bv)}
// ---------------------------------------------------------------------------
__global__ void phosphene_phase1(const float* __restrict__ stim,
                                 const float* __restrict__ phi,
                                 float* __restrict__ ws,
                                 float deg2pix) {
    int idx = blockIdx.x * blockDim.x + threadIdx.x;
    if (idx >= 2048) return;
    int b = idx >> 10;
    int e = idx & 1023;
    const float* ph = phi + b * 13;

    // base grid (meshgrid 'xy', flattened C-order): gx=xc[e%32], gy=xc[e/32]
    float gxb = -15.0f + (30.0f / 31.0f) * (float)(e & 31);
    float gyb = -15.0f + (30.0f / 31.0f) * (float)(e >> 5);

    float theta = ph[2] * 0.017453292519943295f;
    float c = cosf(theta), s = sinf(theta);
    float gx = gxb * c - gyb * s + ph[0] * 3.5f;
    float gy = gxb * s + gyb * c + ph[1] * 3.5f;

    // ewk = exp((gx + i gy)/K), K = 17.3
    float ex  = expf(gx * (1.0f / 17.3f));
    float ang = gy * (1.0f / 17.3f);
    float er = ex * cosf(ang), ei = ex * sinf(ang);

    // z = A*B*(ewk - 1) / (B - A*ewk), A=0.75, B=120, A*B=90
    float nr = 90.0f * (er - 1.0f), ni = 90.0f * ei;
    float dr = 120.0f - 0.75f * er, di = -0.75f * ei;
    float inv = 1.0f / (dr * dr + di * di);
    float vx = (nr * dr + ni * di) * inv;
    float vy = (ni * dr - nr * di) * inv;

    float r = sqrtf(vx * vx + vy * vy);
    float M = 17.3f * (1.0f / (r + 0.75f) - 1.0f / (r + 120.0f));

    float spread   = clampf(ph[3], 0.1f, 10.0f);
    float bright   = clampf(ph[4], 0.1f, 5.0f);
    float sizes    = clampf(ph[5], 0.1f, 5.0f);
    float thr      = clampf(ph[6], 0.1f, 5.0f);
    float contrast = clampf(ph[7], 0.1f, 5.0f);

    float I    = stim[b * 1024 + e] * 8e-5f;
    float Ieff = fmaxf(I - 2.39e-5f * thr, 0.0f);
    float Q    = Ieff * (0.00017f * 300.0f);
    float x    = 19152642.5f * (Q - 1.057e-7f);
    float Bv   = (1.0f / (1.0f + expf(-x))) * bright;
    Bv = powf(Bv, 1.0f / fmaxf(contrast, 0.5f));

    float size_base = sqrtf(I / (0.000675f * spread));
    float sigma     = size_base * (0.5f / (M + 1e-9f)) * sizes;
    float sigma_px  = fmaxf(sigma * deg2pix, 1.0f);
    float k         = 0.5f / (sigma_px * sigma_px);

    float* w = ws + b * 4096;
    w[e]        = k;
    w[1024 + e] = vx * deg2pix;
    w[2048 + e] = vy * deg2pix;
    w[3072 + e] = log2f(Bv);
}

// ---------------------------------------------------------------------------
// Phase 2: each wave32 computes 16 pixels (one row segment) x 1024 electrodes.
// Exponent tile via V_WMMA_F32_16X16X4_F32 (rank-4 quadratic form), then
// raw v_exp_f32 per element, accumulated in the D-shaped accumulator.
// ---------------------------------------------------------------------------
__global__ __launch_bounds__(256) void phosphene_phase2(const float* __restrict__ ws,
                                                        float* __restrict__ out) {
    __shared__ float lds[4096];   // [4][1024]: k, Vx, Vy, log2(Bv)

    const int tid  = threadIdx.x;
    const int wid  = blockIdx.x * 8 + (tid >> 5);   // 8 waves per block
    const int b    = wid >> 12;                     // 4096 wave-tiles per batch
    const int rem  = wid & 4095;
    const int y    = rem >> 4;                      // 0..255
    const int x0   = (rem & 15) << 4;               // 0,16,...,240
    const int lane = tid & 31;
    const int m    = lane & 15;
    const bool hi  = lane >= 16;

    // all 8 waves in a block share batch b (512 blocks per batch)
    const float* src = ws + b * 4096;
    for (int i = tid; i < 4096; i += 256) lds[i] = src[i];
    __syncthreads();

    // pixel-space coordinates: P_i = -128 + (256/255)*i  (FOV-independent)
    const float STEP = 256.0f / 255.0f;
    const float Pp = STEP * ((float)m - 7.5f);            // centered pixel x
    const float cx = -128.0f + STEP * ((float)x0 + 7.5f); // tile center x
    const float cy = -128.0f + STEP * (float)y;           // row y (P'y == 0)

    // branchless half-wave select weights (hoisted)
    const float hf = hi ? 1.0f : 0.0f;
    const float lf = 1.0f - hf;

    // A matrix 16x4 f32: lanes 0-15 -> (K0,K1)=(P'^2, P'); lanes 16-31 -> (0,1)
    v2f a;
    a.x = lf * (Pp * Pp);
    a.y = hi ? 1.0f : Pp;

    v8f cz = {0.0f, 0.0f, 0.0f, 0.0f, 0.0f, 0.0f, 0.0f, 0.0f};
    float acc[8] = {0.0f, 0.0f, 0.0f, 0.0f, 0.0f, 0.0f, 0.0f, 0.0f};

#pragma unroll 4
    for (int t = 0; t < 64; ++t) {
        int e = (t << 4) + m;                 // electrode N = lane&15 of tile t
        float k  = lds[e];
        float Vx = lds[1024 + e];
        float Vy = lds[2048 + e];
        float lB = lds[3072 + e];
        float vxp = Vx - cx;
        float vyp = Vy - cy;
        float kl = k * L2E;

        // B matrix 4x16 f32: lanes 0-15 rows (alpha,beta); lanes 16-31 (0,delta)
        // Branchless blend: both candidates computed in all lanes, combined via
        // fma so the compiler cannot sink work into divergent branches.
        float blo = 2.0f * kl * vxp;                              // beta
        float bhi = fmaf(-kl, fmaf(vxp, vxp, vyp * vyp), lB);     // delta
        v2f bb;
        bb.x = lf * (-kl);                                        // alpha | 0
        bb.y = fmaf(hf, bhi, lf * blo);                           // beta  | delta

        // exponent tile (16 pixels x 16 electrodes), D = A x B + 0
        v8f d = __builtin_amdgcn_wmma_f32_16x16x4_f32(
            false, a, false, bb, (short)0, cz, false, false);

#pragma unroll
        for (int i = 0; i < 8; ++i)
            acc[i] += FAST_EXP2(d[i]);        // exp2 with log2e pre-folded
    }

    // reduce over electrodes: sum the 16 lanes of each half per D-VGPR
#pragma unroll
    for (int i = 0; i < 8; ++i) {
        float v = acc[i];
        v += __shfl_xor(v, 1, 32);
        v += __shfl_xor(v, 2, 32);
        v += __shfl_xor(v, 4, 32);
        v += __shfl_xor(v, 8, 32);
        acc[i] = v;
    }

    // lane 0 holds pixels M=0..7 (acc[0..7]); lane 16 holds M=8..15
    if (m == 0) {
        float* op = out + b * 65536 + y * 256 + x0 + (hi ? 8 : 0);
#pragma unroll
        for (int i = 0; i < 8; ++i)
            op[i] = fminf(fmaxf(2.0f * acc[i], 0.0f), 1.0f);
    }
}

// ---------------------------------------------------------------------------
// Host: replicate numpy float64 _compute_fov()
// ---------------------------------------------------------------------------
static double compute_deg2pix() {
    double maxv = 0.0;
    for (int i = 0; i < 32; ++i) {
        double gy = -15.0 + 30.0 * (double)i / 31.0;
        for (int j = 0; j < 32; ++j) {
            double gx = -15.0 + 30.0 * (double)j / 31.0;
            double re = gx / 17.3, im = gy / 17.3;
            double exd = exp(re);
            double er = exd * cos(im), ei = exd * sin(im);
            double nr = 90.0 * (er - 1.0), ni = 90.0 * ei;
            double dr = 120.0 - 0.75 * er, di = -0.75 * ei;
            double inv = 1.0 / (dr * dr + di * di);
            double zr = (nr * dr + ni * di) * inv;
            double zi = (ni * dr - nr * di) * inv;
            double am = fabs(zr) > fabs(zi) ? fabs(zr) : fabs(zi);
            if (am > maxv) maxv = am;
        }
    }
    double fov = maxv * 1.1;
    return 256.0 / (fov * 2.0);
}

extern "C" void kernel_launch(void* const* d_in, const int* in_sizes, int n_in,
                              void* d_out, int out_size, void* d_ws, size_t ws_size,
                              hipStream_t stream) {
    const float* stim = (const float*)d_in[0];  // (2,1,32,32) f32
    const float* phi  = (const float*)d_in[1];  // (2,13) f32
    float* out = (float*)d_out;                 // (2,1,256,256) f32
    float* ws  = (float*)d_ws;                  // 2*4*1024 f32 = 32 KB

    float deg2pix = (float)compute_deg2pix();

    phosphene_phase1<<<8, 256, 0, stream>>>(stim, phi, ws, deg2pix);
    phosphene_phase2<<<1024, 256, 0, stream>>>(ws, out);
}